// LinearChainCRF_20426864460358
// MI455X (gfx1250) — compile-verified
//
#include <hip/hip_runtime.h>
#include <hip/hip_bf16.h>

#define BB 64
#define SS 2048
#define TT 128
#define IMPOSSIBLE_V (-100000.0f)

typedef float v2f __attribute__((ext_vector_type(2)));
typedef float v8f __attribute__((ext_vector_type(8)));
typedef int   v4i __attribute__((ext_vector_type(4)));

#if defined(__has_builtin)
# if __has_builtin(__builtin_amdgcn_global_load_async_to_lds_b128) && \
     __has_builtin(__builtin_amdgcn_s_wait_asynccnt)
#  define USE_ASYNC_LDS 1
# endif
#endif
#ifndef USE_ASYNC_LDS
# define USE_ASYNC_LDS 0
#endif

// ---------------------------------------------------------------------------
// Forward (partition function) kernel.
// 4 workgroups x 256 threads (8 wave32). WG g owns batch rows [16g, 16g+16).
// Per step: res[b,j] = M_b + log( sum_i exp(alpha[b,i]-M_b) * expTrans[i,j] )
// computed as an f32 GEMM with V_WMMA_F32_16X16X4_F32 (K fully unrolled).
// expTrans tile (128 x 16 cols) lives in 64 VGPRs per lane (step-invariant).
// Emissions are double-buffered LDS tiles filled by GLOBAL_LOAD_ASYNC_TO_LDS
// one step ahead (ASYNCcnt), so the only per-step global traffic is off the
// critical path.  Row maxima via wave shuffles -> only 2 barriers per step.
// ---------------------------------------------------------------------------
__global__ __launch_bounds__(256)
void crf_forward_kernel(const float* __restrict__ emis,   // (B,S,T) f32
                        const int*   __restrict__ mask,   // (B,S)   i32 (bool)
                        const int*   __restrict__ forb,   // (T,T)   i32 (bool)
                        const float* __restrict__ trans,  // (T,T)
                        const float* __restrict__ startT, // (T)
                        const float* __restrict__ endT,   // (T)
                        float*       __restrict__ z_out)  // (B)
{
    __shared__ float s_alpha[16 * TT];        // alpha[16][128]
    __shared__ float s_aexp [16 * TT];        // exp(alpha - rowmax)
#if USE_ASYNC_LDS
    __shared__ float s_em[2][16 * TT];        // double-buffered emission tiles
#endif
    __shared__ float s_rowmax[16];
    __shared__ int   s_mrow[16];

    const int tid   = threadIdx.x;
    const int wg    = blockIdx.x;             // 0..3
    const int base  = wg * 16;                // batch row base
    const int lane  = tid & 31;
    const int wave  = tid >> 5;               // 0..7 -> output tag tile
    const int khalf = lane >> 4;              // 0/1
    const int l15   = lane & 15;
    const int jcol  = wave * 16 + l15;        // this lane's output tag column

    // --- Preload B-matrix tile into registers: breg[kk] = {E[4kk+2h][j], E[4kk+2h+1][j]}
    float2 breg[32];
#pragma unroll
    for (int kk = 0; kk < 32; ++kk) {
        const int i0  = kk * 4 + khalf * 2;
        const int ix0 = i0 * TT + jcol;
        breg[kk].x = forb[ix0]      ? 0.0f : expf(trans[ix0]);
        breg[kk].y = forb[ix0 + TT] ? 0.0f : expf(trans[ix0 + TT]);
    }

    // --- alpha0 = start_transitions + emissions[:,0,:]
    for (int idx = tid; idx < 16 * TT; idx += 256) {
        const int r = idx >> 7, c = idx & 127;
        s_alpha[idx] = startT[c] + emis[(size_t)(base + r) * SS * TT + c];
    }

#if USE_ASYNC_LDS
    // prime the pipeline: async-load emissions for s=1 into buffer 1
    {
        const int r = tid >> 4, c8 = (tid & 15) * 8;
        const float* g = emis + (size_t)(base + r) * SS * TT + (size_t)1 * TT + c8;
        float* l = &s_em[1][r * TT + c8];
        __builtin_amdgcn_global_load_async_to_lds_b128(
            (v4i*)g,       (v4i*)l,       0, 0);
        __builtin_amdgcn_global_load_async_to_lds_b128(
            (v4i*)(g + 4), (v4i*)(l + 4), 0, 0);
    }
#endif
    __syncthreads();

    const float2* A2 = (const float2*)s_aexp; // [16][64] pairs of K

    for (int s = 1; s < SS; ++s) {
        // ---- phase A: per-row max via shuffles, exp-shift, fetch mask ----
        {
            const int row = 2 * wave + khalf;           // 0..15
            const int seg = l15;                        // 0..15 (8 cols each)
            const float* ap = s_alpha + row * TT + seg * 8;
            const float4 x0 = *(const float4*)(ap);
            const float4 x1 = *(const float4*)(ap + 4);
            float mx = fmaxf(fmaxf(fmaxf(x0.x, x0.y), fmaxf(x0.z, x0.w)),
                             fmaxf(fmaxf(x1.x, x1.y), fmaxf(x1.z, x1.w)));
            mx = fmaxf(mx, __shfl_xor(mx, 1));
            mx = fmaxf(mx, __shfl_xor(mx, 2));
            mx = fmaxf(mx, __shfl_xor(mx, 4));
            mx = fmaxf(mx, __shfl_xor(mx, 8));          // max over the 16-lane half
            if (seg == 0) {
                s_rowmax[row] = mx;
                s_mrow[row]   = mask[(size_t)(base + row) * SS + s];
            }
            float4 e0, e1;
            e0.x = __expf(x0.x - mx); e0.y = __expf(x0.y - mx);
            e0.z = __expf(x0.z - mx); e0.w = __expf(x0.w - mx);
            e1.x = __expf(x1.x - mx); e1.y = __expf(x1.y - mx);
            e1.z = __expf(x1.z - mx); e1.w = __expf(x1.w - mx);
            float* xp = s_aexp + row * TT + seg * 8;
            *(float4*)(xp)     = e0;
            *(float4*)(xp + 4) = e1;
        }

#if USE_ASYNC_LDS
        // issue async emission loads for step s+1 (consumed next iteration)
        if (s + 1 < SS) {
            const int r = tid >> 4, c8 = (tid & 15) * 8;
            const float* g = emis + (size_t)(base + r) * SS * TT
                                  + (size_t)(s + 1) * TT + c8;
            float* l = &s_em[(s + 1) & 1][r * TT + c8];
            __builtin_amdgcn_global_load_async_to_lds_b128(
                (v4i*)g,       (v4i*)l,       0, 0);
            __builtin_amdgcn_global_load_async_to_lds_b128(
                (v4i*)(g + 4), (v4i*)(l + 4), 0, 0);
            __builtin_amdgcn_s_wait_asynccnt(2);  // prior step's tile done
        } else {
            __builtin_amdgcn_s_wait_asynccnt(0);
        }
#else
        if (tid < 64 && s + 1 < SS) {
            const int r = tid >> 2, ch = tid & 3;
            __builtin_prefetch(
                emis + (size_t)(base + r) * SS * TT + (size_t)(s + 1) * TT + ch * 32,
                0, 3);
        }
#endif
        __syncthreads();

        // ---- phase B: 16x16 output tile per wave via f32 WMMA, K=128 ----
        {
            v8f acc = {0.f, 0.f, 0.f, 0.f, 0.f, 0.f, 0.f, 0.f};
#pragma unroll
            for (int kk = 0; kk < 32; ++kk) {
                const float2 a2 = A2[l15 * 64 + kk * 2 + khalf];
                v2f av; av.x = a2.x; av.y = a2.y;
                v2f bv; bv.x = breg[kk].x; bv.y = breg[kk].y;
                acc = __builtin_amdgcn_wmma_f32_16x16x4_f32(
                    false, av, false, bv, (short)0, acc, false, false);
            }
            // D layout: vgpr r holds row (r + 8*khalf), col = l15
#pragma unroll
            for (int r = 0; r < 8; ++r) {
                const int row = r + 8 * khalf;
#if USE_ASYNC_LDS
                const float emv = s_em[s & 1][row * TT + jcol];
#else
                const float emv = emis[(size_t)(base + row) * SS * TT
                                       + (size_t)s * TT + jcol];
#endif
                const float val = s_rowmax[row]
                                + __logf(fmaxf(acc[r], 1e-30f)) + emv;
                if (s_mrow[row]) s_alpha[row * TT + jcol] = val;
            }
        }
        __syncthreads();
    }

    // ---- z[b] = logsumexp_j(alpha[b,j] + end_transitions[j]) ----
    if (tid < 16) {
        const int r = tid;
        float mx = -3.4e38f;
        for (int j = 0; j < TT; ++j)
            mx = fmaxf(mx, s_alpha[r * TT + j] + endT[j]);
        float sum = 0.0f;
        for (int j = 0; j < TT; ++j)
            sum += expf(s_alpha[r * TT + j] + endT[j] - mx);
        z_out[base + r] = mx + logf(sum);
    }
}

// ---------------------------------------------------------------------------
// Posterior path score: masked scan == masked sum (contiguous mask), fully
// parallel reduction, one block per batch element.
// ---------------------------------------------------------------------------
__global__ __launch_bounds__(256)
void crf_posterior_kernel(const float* __restrict__ emis,
                          const int*   __restrict__ mask,
                          const int*   __restrict__ tags,
                          const int*   __restrict__ forb,
                          const float* __restrict__ trans,
                          const float* __restrict__ startT,
                          const float* __restrict__ endT,
                          float*       __restrict__ post_out)
{
    __shared__ float s_sum[256];
    __shared__ int   s_cnt[256];
    const int b = blockIdx.x, tid = threadIdx.x;

    float acc = 0.0f;
    int   cnt = 0;
    for (int k = tid; k < SS; k += 256) {
        const int m = mask[(size_t)b * SS + k];
        cnt += (m != 0);
        if (k >= 1 && m) {
            const int tp = tags[(size_t)b * SS + k - 1];
            const int tc = tags[(size_t)b * SS + k];
            const int ti = tp * TT + tc;
            const float tr = forb[ti] ? IMPOSSIBLE_V : trans[ti];
            acc += tr + emis[(size_t)b * SS * TT + (size_t)k * TT + tc];
        }
    }
    s_sum[tid] = acc;
    s_cnt[tid] = cnt;
    __syncthreads();
    for (int off = 128; off > 0; off >>= 1) {
        if (tid < off) {
            s_sum[tid] += s_sum[tid + off];
            s_cnt[tid] += s_cnt[tid + off];
        }
        __syncthreads();
    }
    if (tid == 0) {
        const int len = s_cnt[0];
        const int t0  = tags[(size_t)b * SS];
        const int tl  = tags[(size_t)b * SS + len - 1];
        post_out[b] = startT[t0] + emis[(size_t)b * SS * TT + t0]
                    + s_sum[0] + endT[tl];
    }
}

// ---------------------------------------------------------------------------
// Final scalar: mean over (1,B,B) of post[j]-z[i]  ==  mean(post) - mean(z)
// ---------------------------------------------------------------------------
__global__ __launch_bounds__(64)
void crf_final_kernel(const float* __restrict__ z,
                      const float* __restrict__ post,
                      float*       __restrict__ out)
{
    __shared__ float s_z[64], s_p[64];
    const int t = threadIdx.x;
    s_z[t] = z[t];
    s_p[t] = post[t];
    __syncthreads();
    for (int off = 32; off > 0; off >>= 1) {
        if (t < off) { s_z[t] += s_z[t + off]; s_p[t] += s_p[t + off]; }
        __syncthreads();
    }
    if (t == 0) out[0] = (s_p[0] - s_z[0]) * (1.0f / (float)BB);
}

extern "C" void kernel_launch(void* const* d_in, const int* in_sizes, int n_in,
                              void* d_out, int out_size, void* d_ws, size_t ws_size,
                              hipStream_t stream) {
    (void)in_sizes; (void)n_in; (void)out_size; (void)ws_size;
    const float* emis   = (const float*)d_in[0];  // (B,S,T)
    const int*   mask   = (const int*)  d_in[1];  // (B,S) bool->i32
    const int*   tags   = (const int*)  d_in[2];  // (B,S)
    const int*   forb   = (const int*)  d_in[3];  // (T,T) bool->i32
    const float* trans  = (const float*)d_in[4];  // (T,T)
    const float* startT = (const float*)d_in[5];  // (T)
    const float* endT   = (const float*)d_in[6];  // (T)
    float* out = (float*)d_out;

    float* z    = (float*)d_ws;        // B floats
    float* post = z + BB;              // B floats

    crf_forward_kernel<<<4, 256, 0, stream>>>(emis, mask, forb, trans,
                                              startT, endT, z);
    crf_posterior_kernel<<<BB, 256, 0, stream>>>(emis, mask, tags, forb, trans,
                                                 startT, endT, post);
    crf_final_kernel<<<1, 64, 0, stream>>>(z, post, out);
}